// EMRouting_89060441849994
// MI455X (gfx1250) — compile-verified
//
#include <hip/hip_runtime.h>
#include <math.h>

// ---- Problem constants (fixed by reference) ----
#define B_   16
#define CH_  8192
#define O_   32
#define P_   16
#define EPSF 1e-9f
#define TWO_PI_F 6.28318530717958647692f

// ---- Tiling ----
#define WAVES  8              // 256 threads / wave32
#define CHW    16             // children per wave
#define CHUNK  (WAVES*CHW)    // 128 children per workgroup
#define NCHUNK (CH_/CHUNK)    // 64 chunks per batch element
#define NBUF   3              // async stage depth (triple buffer)
// Sums layout per b: S0[o] (32) | S1[p][o] (512, transposed) | S2[p][o] (512)
#define SUMS_PER_B (O_ + O_*P_ + O_*P_)   // 1056 floats
#define SUMS_FLOATS (B_*SUMS_PER_B)

// ---- CDNA5 async global->LDS path (guarded; falls back to direct loads) ----
#if defined(__HIP_DEVICE_COMPILE__) && __has_builtin(__builtin_amdgcn_global_load_async_to_lds_b128)
#define USE_ASYNC 1
#else
#define USE_ASYNC 0
#endif

#if USE_ASYNC
typedef int vi4 __attribute__((__vector_size__(4 * sizeof(int))));
typedef __attribute__((address_space(1))) vi4 gvi4;   // global
typedef __attribute__((address_space(3))) vi4 lvi4;   // LDS

__device__ __forceinline__ void async_ld_b128(const void* g, void* l) {
  // 16B per lane, memory -> LDS, tracked by ASYNCcnt
  __builtin_amdgcn_global_load_async_to_lds_b128(
      (gvi4*)(unsigned long long)g,
      (lvi4*)(unsigned int)(unsigned long long)l,
      0, 0);
}
__device__ __forceinline__ void wait_async(int keep) {
#if __has_builtin(__builtin_amdgcn_s_wait_asynccnt)
  if (keep >= 8)      __builtin_amdgcn_s_wait_asynccnt(8);
  else if (keep == 4) __builtin_amdgcn_s_wait_asynccnt(4);
  else                __builtin_amdgcn_s_wait_asynccnt(0);
#else
  if (keep >= 8)      asm volatile("s_wait_asynccnt 8" ::: "memory");
  else if (keep == 4) asm volatile("s_wait_asynccnt 4" ::: "memory");
  else                asm volatile("s_wait_asynccnt 0" ::: "memory");
#endif
}
#endif

__device__ __forceinline__ float fast_rcp(float x) {
#if defined(__HIP_DEVICE_COMPILE__) && __has_builtin(__builtin_amdgcn_rcpf)
  return __builtin_amdgcn_rcpf(x);     // v_rcp_f32 (x >= 1 here, safe)
#else
  return 1.0f / x;
#endif
}

// ============================================================================
// Streaming pass: (optional fused E-step) + moment accumulation over CH.
//   EM=false: rr uniform (1/O); params unused.
//   EM=true : logits from previous (mu, w=0.5/var, K=log(a+eps)-Sum(0.5 log 2pi var)),
//             softmax over o (32 lanes), rr' = rr * child_prob.
// Votes streamed via triple-buffered global_load_async_to_lds_b128; reduction
// via conflict-free ds_add_f32 (S1/S2 transposed [p][o]) then global f32 atomics.
// ============================================================================
template<bool EM>
__global__ __launch_bounds__(256) void em_pass(const float* __restrict__ votes,
                                               const float* __restrict__ child_prob,
                                               const float* __restrict__ params,
                                               float* __restrict__ sums)
{
  __shared__ float sacc[SUMS_PER_B];
#if USE_ASYNC
  __shared__ float stage[WAVES][NBUF][O_*P_];   // 2 KB per child, 3-deep per wave
#endif
  const int tid  = threadIdx.x;
  const int w    = tid >> 5;
  const int lane = tid & 31;                 // lane == parent capsule o
  const int bi   = blockIdx.x;
  const int b    = bi / NCHUNK;
  const int ch0  = (bi % NCHUNK) * CHUNK + w * CHW;

  for (int i = tid; i < SUMS_PER_B; i += blockDim.x) sacc[i] = 0.f;
  __syncthreads();

  // Per-lane (per-o) parameters from previous iteration's stats.
  float mu[P_], wgt[P_], K = 0.f;
  if (EM) {
    const float* pb = params + (size_t)b * SUMS_PER_B;   // [mu 512][w 512][K 32], [o][p]
#pragma unroll
    for (int p = 0; p < P_; ++p) mu[p]  = pb[lane*P_ + p];
#pragma unroll
    for (int p = 0; p < P_; ++p) wgt[p] = pb[O_*P_ + lane*P_ + p];
    K = pb[2*O_*P_ + lane];
  }

  float s0 = 0.f, s1[P_], s2[P_];
#pragma unroll
  for (int p = 0; p < P_; ++p) { s1[p] = 0.f; s2[p] = 0.f; }

  const float* vbase = votes + ((size_t)b * CH_ + ch0) * (O_*P_);

#if USE_ASYNC
  // Prologue: stage children 0..NBUF-1 (4 x b128 per child per lane).
#pragma unroll
  for (int k = 0; k < NBUF; ++k) {
    const float* g = vbase + (size_t)k * (O_*P_);
    float* l = &stage[w][k][0];
#pragma unroll
    for (int j = 0; j < 4; ++j) async_ld_b128(g + j*128 + lane*4, l + j*128 + lane*4);
  }
  int bu = 0;                                // stage buffer for child i (== i % NBUF)
#endif

  for (int i = 0; i < CHW; ++i) {
    const float cpv = child_prob[(size_t)b*CH_ + ch0 + i] * (EM ? 1.0f : (1.0f/(float)O_));

    float v[P_];
#if USE_ASYNC
    // Outstanding = stages i..i+2 (4 ops each); keep newest, force oldest done.
    wait_async(4 * (CHW-1-i < NBUF-1 ? CHW-1-i : NBUF-1));
    const float4* vp = (const float4*)&stage[w][bu][lane*P_];
#else
    const float4* vp = (const float4*)(vbase + (size_t)i*(O_*P_) + lane*P_);
#endif
    { float4 a0 = vp[0], a1 = vp[1], a2 = vp[2], a3 = vp[3];
      *(float4*)&v[0] = a0; *(float4*)&v[4] = a1; *(float4*)&v[8] = a2; *(float4*)&v[12] = a3; }

    float rp;
    if (EM) {
      // logit[o] = log(a+eps) - Sum_p 0.5*log(2pi*var) - Sum_p (v-mu)^2 * (0.5/var)
      float t = 0.f;
#pragma unroll
      for (int p = 0; p < P_; ++p) { float d = v[p] - mu[p]; t = fmaf(d*d, wgt[p], t); }
      float logit = K - t;
      // softmax over o == over the 32 lanes of this wave
      float m = logit;
#pragma unroll
      for (int off = 16; off >= 1; off >>= 1) m = fmaxf(m, __shfl_xor(m, off, 32));
      float e = __expf(logit - m);
      float ssum = e;
#pragma unroll
      for (int off = 16; off >= 1; off >>= 1) ssum += __shfl_xor(ssum, off, 32);
      rp = e * fast_rcp(ssum) * cpv;         // ssum >= 1 (max lane contributes 1)
    } else {
      rp = cpv;                              // uniform routing, 1/O folded into cpv
    }

    s0 += rp;
#pragma unroll
    for (int p = 0; p < P_; ++p) {
      s1[p] = fmaf(rp, v[p],        s1[p]);
      s2[p] = fmaf(rp * v[p], v[p], s2[p]);
    }

#if USE_ASYNC
    if (i + NBUF < CHW) {                    // refill the buffer just consumed
      const float* g = vbase + (size_t)(i + NBUF) * (O_*P_);
      float* l = &stage[w][bu][0];
#pragma unroll
      for (int j = 0; j < 4; ++j) async_ld_b128(g + j*128 + lane*4, l + j*128 + lane*4);
    }
    bu = (bu == NBUF-1) ? 0 : bu + 1;
#endif
  }

  // Cross-wave reduction. S1/S2 stored [p][o] so the 32 lanes (o) hit 32
  // consecutive LDS words per ds_add_f32 -> conflict-free.
  atomicAdd(&sacc[lane], s0);
#pragma unroll
  for (int p = 0; p < P_; ++p) atomicAdd(&sacc[O_ + p*O_ + lane], s1[p]);
#pragma unroll
  for (int p = 0; p < P_; ++p) atomicAdd(&sacc[O_ + O_*P_ + p*O_ + lane], s2[p]);
  __syncthreads();

  float* gs = sums + (size_t)b * SUMS_PER_B;
  for (int i = tid; i < SUMS_PER_B; i += blockDim.x) atomicAdd(&gs[i], sacc[i]);
}

// ============================================================================
// Stats kernel: sums -> (mu, var) -> a_out; writes either next-iteration
// params [mu|0.5/var|K] ([o][p] layout) or the final packed output [B, O, P+1].
// One wave per batch element, lane == o. Sums S1/S2 are [p][o].
// ============================================================================
__global__ __launch_bounds__(32) void em_stats(const float* __restrict__ sums,
                                               const float* __restrict__ beta_a,
                                               const float* __restrict__ beta_v,
                                               float inv_temp, int final_out,
                                               float* __restrict__ out)
{
  const int b = blockIdx.x, o = threadIdx.x;
  const float* sb = sums + (size_t)b * SUMS_PER_B;
  const float S0 = sb[o];
  const float denom = S0 + EPSF;
  const float bv = beta_v[o], ba = beta_a[o];

  float mu[P_], var[P_];
  float cost = 0.f, csum = 0.f;
#pragma unroll
  for (int p = 0; p < P_; ++p) {
    const float S1 = sb[O_ + p*O_ + o];
    const float S2 = sb[O_ + O_*P_ + p*O_ + o];
    const float m  = S1 / denom;
    const float vn = S2 - 2.f*m*S1 + m*m*S0;       // == Sum rr'(v-mu)^2
    const float vv = fmaxf(vn, 0.f) / denom + EPSF;
    mu[p] = m; var[p] = vv;
    cost += (bv + 0.5f*logf(vv)) * S0;
    csum += 0.5f*logf(TWO_PI_F * vv);
  }
  const float a = 1.f / (1.f + __expf(-inv_temp * (ba - cost)));

  if (final_out) {
    float* ob = out + ((size_t)b*O_ + o) * (P_ + 1);
#pragma unroll
    for (int p = 0; p < P_; ++p) ob[p] = mu[p];
    ob[P_] = a;
  } else {
    float* pb = out + (size_t)b * SUMS_PER_B;
#pragma unroll
    for (int p = 0; p < P_; ++p) pb[o*P_ + p] = mu[p];
#pragma unroll
    for (int p = 0; p < P_; ++p) pb[O_*P_ + o*P_ + p] = 0.5f / var[p];
    pb[2*O_*P_ + o] = logf(a + EPSF) - csum;       // softmax-logit constant
  }
}

__global__ void em_zero(float* __restrict__ p, int n) {
  int i = blockIdx.x * blockDim.x + threadIdx.x;
  if (i < n) p[i] = 0.f;
}

// ============================================================================
// Launch: 3 streaming passes over votes (roofline ~33us @ 23.3 TB/s) + tiny
// stats kernels between them. All on `stream`; ws holds two ping-pong sum
// regions and one params region (zeroed/overwritten every call).
// ============================================================================
extern "C" void kernel_launch(void* const* d_in, const int* in_sizes, int n_in,
                              void* d_out, int out_size, void* d_ws, size_t ws_size,
                              hipStream_t stream) {
  const float* votes = (const float*)d_in[0];
  const float* cp    = (const float*)d_in[1];
  const float* ba    = (const float*)d_in[2];
  const float* bv    = (const float*)d_in[3];
  float* out = (float*)d_out;
  float* ws  = (float*)d_ws;

  float* R0 = ws;
  float* R1 = ws + SUMS_FLOATS;
  float* PR = ws + 2*SUMS_FLOATS;

  const dim3 pg(B_ * NCHUNK), pb(256);
  const dim3 zg((SUMS_FLOATS + 255) / 256), zb(256);

  const float it0 = 0.01f * (1.f - 0.95f);
  const float it1 = 0.01f * (1.f - 0.95f*0.95f);
  const float it2 = 0.01f * (1.f - 0.95f*0.95f*0.95f);

  // it 0: uniform routing M-step
  em_zero<<<zg, zb, 0, stream>>>(R0, SUMS_FLOATS);
  em_pass<false><<<pg, pb, 0, stream>>>(votes, cp, nullptr, R0);
  em_stats<<<B_, 32, 0, stream>>>(R0, ba, bv, it0, 0, PR);
  // it 1: fused E-step(0) + M-step(1)
  em_zero<<<zg, zb, 0, stream>>>(R1, SUMS_FLOATS);
  em_pass<true><<<pg, pb, 0, stream>>>(votes, cp, PR, R1);
  em_stats<<<B_, 32, 0, stream>>>(R1, ba, bv, it1, 0, PR);
  // it 2: fused E-step(1) + M-step(2), final output
  em_zero<<<zg, zb, 0, stream>>>(R0, SUMS_FLOATS);
  em_pass<true><<<pg, pb, 0, stream>>>(votes, cp, PR, R0);
  em_stats<<<B_, 32, 0, stream>>>(R0, ba, bv, it2, 1, out);
}